// BEVPoolOp_9371618640642
// MI455X (gfx1250) — compile-verified
//
#include <hip/hip_runtime.h>
#include <stdint.h>

#define N_     6
#define C_     80
#define DHW_   332288            // 118*32*88
#define CDHW_  (26583040)        // C_*DHW_
#define M_     1036800
#define K_     129600
#define L_     8
#define IVB    16                // intervals per block
#define PTS    (IVB * L_)        // 128 points per block

#if __has_builtin(__builtin_amdgcn_tensor_load_to_lds) && __has_builtin(__builtin_amdgcn_s_wait_tensorcnt)
#define USE_TDM 1
#else
#define USE_TDM 0
#endif

typedef unsigned int v4u __attribute__((ext_vector_type(4)));
typedef int          v4i __attribute__((ext_vector_type(4)));
typedef int          v8i __attribute__((ext_vector_type(8)));
typedef float        v2f __attribute__((ext_vector_type(2)));
typedef float        v8f __attribute__((ext_vector_type(8)));

__global__ __launch_bounds__(256) void bevpool_kernel(
    const float* __restrict__ cam, const float* __restrict__ dw,
    const int* __restrict__ idxs, const int* __restrict__ itv,
    float* __restrict__ out)
{
    __shared__ float    featLds[PTS * C_];   // 40 KB: per-point 80-ch vectors
    __shared__ float    pooled[IVB * 81];    // stride 81 -> conflict-free transpose
    __shared__ float    lds_w[PTS];
    __shared__ uint64_t lds_ga[PTS];

    const int tid = threadIdx.x;
    const int r0  = blockIdx.x * IVB;

    __builtin_prefetch(&idxs[r0 * L_], 0, 1);   // global_prefetch_b8

    // ---- Phase 1: stage weight + global feature base address per point ----
    if (tid < PTS) {
        const int iv = tid >> 3, p = tid & 7;
        const int r  = r0 + iv;
        const int s  = itv[r * 3 + 0];
        const int e  = itv[r * 3 + 1];
        const int pt = s + p;
        float    wgt = 0.0f;
        uint64_t ga  = (uint64_t)(uintptr_t)cam;
        if (pt < e) {
            const int id  = idxs[pt];
            wgt           = dw[id];
            const int n   = id / DHW_;
            const int rem = id - n * DHW_;
            ga += 4ull * ((uint64_t)n * (uint64_t)CDHW_ + (uint64_t)rem);
        }
        lds_w[tid]  = wgt;
        lds_ga[tid] = ga;
    }
    __syncthreads();

    const int wv   = tid >> 5;
    const int lane = tid & 31;

#if USE_TDM
    // ---- Phase 1b: each wave DMAs 16 per-point channel-columns via TDM ----
    {
        // flat LDS pointers carry LDS byte offset in addr[31:0]
        const uint32_t featBase = (uint32_t)(uintptr_t)(void*)&featLds[0];
        // group1: data_size=4B, tensor_dim0=1, tensor_dim1=80, tile=1x80, stride0=DHW
        const v8i g1 = { (int)(2u << 16), (int)(1u << 16), (int)(80u << 16),
                         (int)(1u << 16), 80, (int)DHW_, 0, 0 };
        const v4i gz4 = { 0, 0, 0, 0 };
        const v8i gz8 = { 0, 0, 0, 0, 0, 0, 0, 0 };
        const int s0 = wv * (PTS / 8);
#pragma unroll
        for (int i = 0; i < PTS / 8; ++i) {
            const int      slot  = s0 + i;
            const uint64_t ga    = lds_ga[slot];
            const uint32_t glo   = (uint32_t)__builtin_amdgcn_readfirstlane((int)(uint32_t)ga);
            const uint32_t ghi   = (uint32_t)__builtin_amdgcn_readfirstlane((int)(uint32_t)(ga >> 32));
            const uint32_t laddr = featBase + (uint32_t)slot * (C_ * 4);
            // group0: count=1 | lds_addr | global_addr[31:0] | global_addr[56:32] + type=2
            v4u g0 = { 1u, laddr, glo, (ghi & 0x01FFFFFFu) | 0x80000000u };
            __builtin_amdgcn_tensor_load_to_lds(g0, g1, gz4, gz4, gz8, 0);
        }
        __builtin_amdgcn_s_wait_tensorcnt(0);
    }
    __syncthreads();

    // ---- Phase 2: pooled(16x80) = A(16x128, block-diag weights) x F(128x80) via
    //      exact f32 WMMA, K swept in 32 slices of 4 points. Waves 0..4 own one
    //      16-channel tile each. EXEC stays all-1s (selects, no branches). ----
    if (wv < 5) {
        const int c0   = wv * 16;
        const int half = lane >> 4;          // K-half: 0 -> K=0,1 ; 1 -> K=2,3
        const int ln   = lane & 15;          // M for A-lanes / N for B-lanes
        v8f acc = {};
#pragma unroll
        for (int s = 0; s < 32; ++s) {
            const int   m  = s >> 1;         // interval owning this K-slice
            const float w0 = lds_w[4 * s + 2 * half + 0];
            const float w1 = lds_w[4 * s + 2 * half + 1];
            const bool  sel = (ln == m);
            v2f A; A.x = sel ? w0 : 0.0f; A.y = sel ? w1 : 0.0f;
            const int pr = 4 * s + 2 * half; // feature rows for this lane half
            v2f B; B.x = featLds[pr * C_ + c0 + ln];
                   B.y = featLds[(pr + 1) * C_ + c0 + ln];
            acc = __builtin_amdgcn_wmma_f32_16x16x4_f32(
                      false, A, false, B, (short)0, acc, false, false);
        }
        // D layout: VGPR j -> M = j + 8*half, N = ln
#pragma unroll
        for (int j = 0; j < 8; ++j) {
            pooled[(j + 8 * half) * 81 + c0 + ln] = acc[j];
        }
    }
#else
    // ---- Fallback: direct global gathers + VALU reduction ----
    for (int ii = 0; ii < IVB / 8; ++ii) {
        const int iv = wv * (IVB / 8) + ii;
        float a0 = 0.f, a1 = 0.f, a2 = 0.f;
        for (int p = 0; p < L_; ++p) {
            const float  wgt = lds_w[iv * L_ + p];
            const float* f   = (const float*)(uintptr_t)lds_ga[iv * L_ + p];
            a0 += wgt * f[(size_t)lane * DHW_];
            a1 += wgt * f[(size_t)(lane + 32) * DHW_];
            if (lane < 16) a2 += wgt * f[(size_t)(lane + 64) * DHW_];
        }
        pooled[iv * 81 + lane]      = a0;
        pooled[iv * 81 + lane + 32] = a1;
        if (lane < 16) pooled[iv * 81 + lane + 64] = a2;
    }
#endif
    __syncthreads();

    // ---- Phase 3: coalesced stores (bev fastest in lane) ----
    {
        const int bevl = tid & (IVB - 1);
        const int cg   = tid / IVB;                  // 0..15
        const int bev  = itv[(r0 + bevl) * 3 + 2];
#pragma unroll
        for (int j = 0; j < C_ / 16; ++j) {          // 5 channels per thread
            const int c = cg * 5 + j;
            out[(size_t)c * K_ + (size_t)bev] = pooled[bevl * 81 + c];
        }
    }
}

extern "C" void kernel_launch(void* const* d_in, const int* in_sizes, int n_in,
                              void* d_out, int out_size, void* d_ws, size_t ws_size,
                              hipStream_t stream)
{
    (void)in_sizes; (void)n_in; (void)out_size; (void)d_ws; (void)ws_size;
    const float* cam  = (const float*)d_in[0];
    const float* dw   = (const float*)d_in[1];
    const int*   idxs = (const int*)d_in[2];
    const int*   itv  = (const int*)d_in[3];
    float*       out  = (float*)d_out;

    dim3 grid(K_ / IVB);   // 8100 blocks
    dim3 block(256);
    bevpool_kernel<<<grid, block, 0, stream>>>(cam, dw, idxs, itv, out);
}